// LinearSuperposition_2147483648613
// MI455X (gfx1250) — compile-verified
//
#include <hip/hip_runtime.h>
#include <hip/hip_bf16.h>
#include <math.h>
#include <stdint.h>

// Problem constants (from reference setup_inputs)
#define BB   4
#define PP   40
#define FF   300
#define FP   304          // F padded to 19 tiles of 16
#define NFT  19           // f-tiles
#define K2P  80           // 2*P, GEMM K dimension
#define WH   16384        // W*H, GEMM N dimension
#define PCOL 256          // panel columns per block
#define EPSF 1e-8f
#define F32_EPS 1.1920929e-07f

typedef float v2f __attribute__((ext_vector_type(2)));
typedef float v4f __attribute__((ext_vector_type(4)));
typedef float v8f __attribute__((ext_vector_type(8)));

// ---------------------------------------------------------------- init
__global__ void init_msum_kernel(float* msum, int n) {
    int i = blockIdx.x * blockDim.x + threadIdx.x;
    if (i < n) msum[i] = 0.0f;
}

// ------------------------------------------------- per-(b,p) mode norms
__global__ void norm_kernel(const float* __restrict__ shapes, float* __restrict__ scale) {
    const int bp = blockIdx.x;
    const int b = bp / PP, p = bp % PP;
    const v4f* s0 = (const v4f*)(shapes + (((size_t)b * 2 + 0) * PP + p) * WH);
    const v4f* s1 = (const v4f*)(shapes + (((size_t)b * 2 + 1) * PP + p) * WH);
    float acc = 0.0f;
    for (int i = threadIdx.x; i < WH / 4; i += 256) {
        v4f a = s0[i], c = s1[i];
        acc += a.x * a.x + a.y * a.y + a.z * a.z + a.w * a.w;
        acc += c.x * c.x + c.y * c.y + c.z * c.z + c.w * c.w;
    }
    __shared__ float red[256];
    red[threadIdx.x] = acc;
    __syncthreads();
    for (int s = 128; s > 0; s >>= 1) {
        if (threadIdx.x < s) red[threadIdx.x] += red[threadIdx.x + s];
        __syncthreads();
    }
    if (threadIdx.x == 0)
        scale[bp] = (1.0f / (float)PP) / (sqrtf(red[0]) + EPSF);
}

// ---------------------------------- build scaled, sign-folded A matrices
__global__ void prep_kernel(const float* __restrict__ mr, const float* __restrict__ scale,
                            float* __restrict__ Ar, float* __restrict__ Ai) {
    int idx = blockIdx.x * blockDim.x + threadIdx.x;
    if (idx >= BB * FP * K2P) return;
    int k = idx % K2P;
    int r = (idx / K2P) % FP;
    int b = idx / (K2P * FP);
    float ar = 0.0f, ai = 0.0f;
    if (r < FF) {
        int p = (k < PP) ? k : (k - PP);
        float sc = scale[b * PP + p];
        float tr = mr[(((size_t)b * FF + r) * 2 + 0) * PP + p];
        float ti = mr[(((size_t)b * FF + r) * 2 + 1) * PP + p];
        if (k < PP) { ar = tr * sc; ai = ti * sc; }
        else        { ar = -ti * sc; ai = tr * sc; }
    }
    Ar[idx] = ar;
    Ai[idx] = ai;
}

// --------------------------------------------------------- aux scalar
__global__ void aux_kernel(const float* __restrict__ mr, float* __restrict__ out_aux) {
    __shared__ float red[256];
    int t = threadIdx.x;
    float v = 0.0f;
    if (t < BB * PP) {
        int b = t / PP, p = t % PP;
        const float* base = mr + (size_t)b * FF * 2 * PP + p;
        float s = 0.0f;
        for (int i = 0; i < FF * 2; ++i) {
            float x = base[(size_t)i * PP];
            s += x * x;
        }
        v = __logf(sqrtf(s) + 1.0f);
    }
    red[t] = v;
    __syncthreads();
    for (int s = 128; s > 0; s >>= 1) {
        if (t < s) red[t] += red[t + s];
        __syncthreads();
    }
    if (t == 0) *out_aux = red[0] / (float)(BB * PP);
}

// ------------------------------------------- async B-panel staging to LDS
// Stage 80 rows x 256 cols (80 KB) of shapes into LDS with
// GLOBAL_LOAD_ASYNC_TO_LDS_B128 (ASYNCcnt-tracked, no VGPR round trip).
__device__ __forceinline__ void fill_panel(float* panel,
                                           const float* __restrict__ gbase, // shapes+b*80*WH+colBase
                                           int tid) {
#pragma unroll
    for (int i = 0; i < 40; ++i) {
        const int chunk = i * 128 + tid;          // 0..5119
        const int row = chunk >> 6;               // 0..79
        const int cg  = (chunk & 63) << 2;        // column 0..252 step 4
        // flat shared-pointer low 32 bits == LDS offset (LDS aperture truncation)
        const unsigned lds_off = (unsigned)(uintptr_t)(panel + row * PCOL + cg);
        const float* ga = gbase + (size_t)row * WH + cg;
        asm volatile("global_load_async_to_lds_b128 %0, %1, off"
                     :: "v"(lds_off), "v"(ga) : "memory");
    }
    asm volatile("s_wait_asynccnt 0x0" ::: "memory");
    __syncthreads();
}

// ------------------------------------------------------ WMMA GEMM core
// One wave: 16(f) x 64(n) output as 4 interleaved 16x16 tiles; tile t covers
// columns wbase + 4*lane16 + t. B panel reads are f-loop-invariant (compiler
// keeps them register-resident); A fragments stream from L2 and are manually
// software-pipelined two k-steps ahead so waits land on 2-step-old loads.
// K=80 in 20 steps of V_WMMA_F32_16X16X4_F32.
// A layout (ISA 7.12.2, 32-bit A 16x4): lanes 0-15 hold K={0,1}, lanes 16-31 K={2,3}.
// C/D: lane(0-15) reg j -> (M=j, N=lane); lane(16-31) reg j -> (M=8+j, N=lane-16).
__device__ __forceinline__ void gemm4_lds(const float* __restrict__ arow,
                                          const float* __restrict__ irow,
                                          const float* lpanel,   // panel + wid*64 + 4*lrow
                                          int lhalf,
                                          v8f (&accR)[4], v8f (&accI)[4]) {
    v2f abuf[2], ibuf[2];
    abuf[0] = *(const v2f*)(arow + 0 + 2 * lhalf);
    ibuf[0] = *(const v2f*)(irow + 0 + 2 * lhalf);
    abuf[1] = *(const v2f*)(arow + 4 + 2 * lhalf);
    ibuf[1] = *(const v2f*)(irow + 4 + 2 * lhalf);
#pragma unroll
    for (int ks = 0; ks < 20; ++ks) {
        const int buf = ks & 1;
        const v2f a_r = abuf[buf];
        const v2f a_i = ibuf[buf];
        if (ks + 2 < 20) {                    // prefetch A two k-steps ahead
            const int ko = (ks + 2) * 4 + 2 * lhalf;
            abuf[buf] = *(const v2f*)(arow + ko);
            ibuf[buf] = *(const v2f*)(irow + ko);
        }
        const int kr = ks * 4 + 2 * lhalf;
        const v4f b0 = *(const v4f*)(lpanel + kr * PCOL);
        const v4f b1 = *(const v4f*)(lpanel + (kr + 1) * PCOL);
#pragma unroll
        for (int t = 0; t < 4; ++t) {
            v2f bv;
            bv.x = b0[t];
            bv.y = b1[t];
            accR[t] = __builtin_amdgcn_wmma_f32_16x16x4_f32(false, a_r, false, bv,
                                                            (short)0, accR[t], false, false);
            accI[t] = __builtin_amdgcn_wmma_f32_16x16x4_f32(false, a_i, false, bv,
                                                            (short)0, accI[t], false, false);
        }
    }
}

// ------------------------------------ pass 1: ln_mag + per-(b,f) mag2 sums
// grid (64, 1, B), block 128 (4 waves). Panel staged once, 19 f-tiles looped.
__global__ __launch_bounds__(128) void pass1_kernel(
        const float* __restrict__ shapes,
        const float* __restrict__ Ar, const float* __restrict__ Ai,
        const int* __restrict__ freqs, const float* __restrict__ bias_table,
        float* __restrict__ ln_mag, float* __restrict__ msum) {
    __shared__ float panel[K2P * PCOL];
    const int tid   = threadIdx.x;
    const int lane  = tid & 31;
    const int wid   = tid >> 5;
    const int lrow  = lane & 15;
    const int lhalf = lane >> 4;
    const int b       = blockIdx.z;
    const int colBase = blockIdx.x * PCOL;

    fill_panel(panel, shapes + (size_t)b * K2P * WH + colBase, tid);

    const int col0 = colBase + wid * 64 + 4 * lrow;
    const float* lpanel = panel + wid * 64 + 4 * lrow;

    for (int ft = 0; ft < NFT; ++ft) {
        const int f0 = ft * 16;
        const float* arow = Ar + ((size_t)b * FP + (f0 + lrow)) * K2P;
        const float* irow = Ai + ((size_t)b * FP + (f0 + lrow)) * K2P;

        v8f accR[4], accI[4];
#pragma unroll
        for (int t = 0; t < 4; ++t) {
            accR[t] = (v8f){0.f, 0.f, 0.f, 0.f, 0.f, 0.f, 0.f, 0.f};
            accI[t] = (v8f){0.f, 0.f, 0.f, 0.f, 0.f, 0.f, 0.f, 0.f};
        }
        gemm4_lds(arow, irow, lpanel, lhalf, accR, accI);

        float rs[8];
#pragma unroll
        for (int j = 0; j < 8; ++j) {
            const int f = f0 + j + 8 * lhalf;
            float bias = 0.0f;
            if (f < FF) bias = bias_table[freqs[b * FF + f]];
            v4f lm;
            float s = 0.0f;
#pragma unroll
            for (int t = 0; t < 4; ++t) {
                float dr = accR[t][j], di = accI[t][j];
                float m2 = dr * dr + di * di;
                s += m2;
                lm[t] = (__logf(m2 + EPSF) + bias) * 0.25f;
            }
            rs[j] = s;
            if (f < FF)
                __builtin_nontemporal_store(lm, (v4f*)(ln_mag + ((size_t)b * FF + f) * WH + col0));
        }
#pragma unroll
        for (int j = 0; j < 8; ++j) {
            float v = rs[j];
            v += __shfl_xor(v, 1);
            v += __shfl_xor(v, 2);
            v += __shfl_xor(v, 4);
            v += __shfl_xor(v, 8);
            const int f = f0 + j + 8 * lhalf;
            if (lrow == 0 && f < FF) atomicAdd(&msum[b * FF + f], v);
        }
    }
}

// --------------------------- pass 2: recompute GEMM (shapes L2-resident),
// divide by sqrt(clip(msum)), write normed_field (B,F,2,W,H)
__global__ __launch_bounds__(128) void pass2_kernel(
        const float* __restrict__ shapes,
        const float* __restrict__ Ar, const float* __restrict__ Ai,
        const float* __restrict__ msum, float* __restrict__ field) {
    __shared__ float panel[K2P * PCOL];
    const int tid   = threadIdx.x;
    const int lane  = tid & 31;
    const int wid   = tid >> 5;
    const int lrow  = lane & 15;
    const int lhalf = lane >> 4;
    const int b       = blockIdx.z;
    const int colBase = blockIdx.x * PCOL;

    fill_panel(panel, shapes + (size_t)b * K2P * WH + colBase, tid);

    const int col0 = colBase + wid * 64 + 4 * lrow;
    const float* lpanel = panel + wid * 64 + 4 * lrow;

    for (int ft = 0; ft < NFT; ++ft) {
        const int f0 = ft * 16;
        const float* arow = Ar + ((size_t)b * FP + (f0 + lrow)) * K2P;
        const float* irow = Ai + ((size_t)b * FP + (f0 + lrow)) * K2P;

        v8f accR[4], accI[4];
#pragma unroll
        for (int t = 0; t < 4; ++t) {
            accR[t] = (v8f){0.f, 0.f, 0.f, 0.f, 0.f, 0.f, 0.f, 0.f};
            accI[t] = (v8f){0.f, 0.f, 0.f, 0.f, 0.f, 0.f, 0.f, 0.f};
        }
        gemm4_lds(arow, irow, lpanel, lhalf, accR, accI);

#pragma unroll
        for (int j = 0; j < 8; ++j) {
            const int f = f0 + j + 8 * lhalf;
            if (f < FF) {
                const float inv = 1.0f / sqrtf(fmaxf(msum[b * FF + f], F32_EPS));
                v4f vr, vi;
#pragma unroll
                for (int t = 0; t < 4; ++t) {
                    vr[t] = accR[t][j] * inv;
                    vi[t] = accI[t][j] * inv;
                }
                const size_t base = (((size_t)b * FF + f) * 2) * WH + col0;
                __builtin_nontemporal_store(vr, (v4f*)(field + base));
                __builtin_nontemporal_store(vi, (v4f*)(field + base + WH));
            }
        }
    }
}

extern "C" void kernel_launch(void* const* d_in, const int* in_sizes, int n_in,
                              void* d_out, int out_size, void* d_ws, size_t ws_size,
                              hipStream_t stream) {
    const float* shapes = (const float*)d_in[0];  // (B,2,P,W,H)
    const float* mr     = (const float*)d_in[1];  // (B,F,2,P)
    const int*   freqs  = (const int*)d_in[2];    // (B,F)
    const float* bias   = (const float*)d_in[3];  // (NBIAS,)
    float* out = (float*)d_out;

    // d_out layout: ln_mag (B,F,WH) | normed_field (B,F,2,WH) | aux (1)
    float* ln_mag = out;
    float* field  = out + (size_t)BB * FF * WH;
    float* aux    = out + (size_t)BB * FF * WH + (size_t)BB * FF * 2 * WH;

    // workspace layout (floats): scale(160) | msum(1200) | Ar(B*FP*K2P) | Ai(B*FP*K2P)
    float* ws    = (float*)d_ws;
    float* scale = ws;
    float* msum  = ws + 160;
    float* Ar    = ws + 1360;
    float* Ai    = ws + 1360 + (size_t)BB * FP * K2P;

    init_msum_kernel<<<(BB * FF + 255) / 256, 256, 0, stream>>>(msum, BB * FF);
    norm_kernel<<<BB * PP, 256, 0, stream>>>(shapes, scale);
    prep_kernel<<<(BB * FP * K2P + 255) / 256, 256, 0, stream>>>(mr, scale, Ar, Ai);
    aux_kernel<<<1, 256, 0, stream>>>(mr, aux);

    dim3 grid(WH / PCOL, 1, BB);   // (64 column strips, 1, batch)
    pass1_kernel<<<grid, 128, 0, stream>>>(shapes, Ar, Ai, freqs, bias, ln_mag, msum);
    pass2_kernel<<<grid, 128, 0, stream>>>(shapes, Ar, Ai, msum, field);
}